// VoxelHashTableFlowTraverse_16887811408407
// MI455X (gfx1250) — compile-verified
//
#include <hip/hip_runtime.h>
#include <stdint.h>

// ---------------------------------------------------------------------------
// VoxelHashTableFlowTraverse: 8-corner hash-grid gather + trilinear blend.
//
// MI455X reasoning:
//  * Feature table (285,696 x 120 f32 = 137 MB) + hash table (4 MB) fit in the
//    192 MB L2 -> gathers are L2-resident; HBM traffic ~= the 480 MB output.
//  * Output is write-once -> NON-TEMPORAL stores so the 480 MB stream does not
//    evict the L2-resident feature table (this is the key perf lever).
//  * WMMA is inapplicable: B-operand rows are gathered per query (no reuse
//    across a 16-row tile). CDNA5-specific paths used instead: async-to-LDS
//    staging (ASYNCcnt) + NT cache-policy b128 stores.
//  * wave32: one wave blends one query; lanes 0..29 each own one float4 chunk
//    of the 480-byte feature row (480 = 30*16, so all b128 accesses aligned).
// ---------------------------------------------------------------------------

#define QPB        32            // queries per block
#define NTHREADS   256           // 8 wave32
#define CH         30            // float4 chunks per 120-float row
#define TABLE_MASK 0xFFFFFu      // 1<<20 table, power of two

typedef float v4f __attribute__((ext_vector_type(4)));

#if defined(__gfx1250__) && __has_builtin(__builtin_amdgcn_global_load_async_to_lds_b32)
#define USE_ASYNC 1
#else
#define USE_ASYNC 0
#endif

__global__ __launch_bounds__(NTHREADS)
void voxel_trilinear_kernel(const float* __restrict__ qp,
                            const float* __restrict__ feat,
                            const int*   __restrict__ table,
                            float*       __restrict__ out,
                            int M)
{
    __shared__ float spts[QPB * 3];
    __shared__ int   sbase[QPB][4];
    __shared__ float sfrac[QPB][4];
    __shared__ int2  sent[QPB][8];   // {voxel index, weight-as-bits} per (q, corner)

    const int tid = threadIdx.x;
    const int qb  = blockIdx.x * QPB;

    // ---- Phase 0: async-stage 32 query points (96 dwords) into LDS ----
    if (tid < QPB * 3) {
        const int g = qb * 3 + tid;
        if (g < M * 3) {
#if USE_ASYNC
            __builtin_amdgcn_global_load_async_to_lds_b32(
                (__attribute__((address_space(1))) int*)(qp + g),
                (__attribute__((address_space(3))) int*)(&spts[tid]),
                0, 0);
#else
            spts[tid] = qp[g];
#endif
        } else {
            spts[tid] = 0.0f;
        }
    }
#if USE_ASYNC
#if __has_builtin(__builtin_amdgcn_s_wait_asynccnt)
    __builtin_amdgcn_s_wait_asynccnt(0);
#else
    asm volatile("s_wait_asynccnt 0" ::: "memory");
#endif
#endif
    __syncthreads();

    // ---- Phase 1a: one thread per query: correctly-rounded div + floor ----
    if (tid < QPB) {
        const float gx = spts[tid * 3 + 0] / 0.1f;   // match reference: p / RES
        const float gy = spts[tid * 3 + 1] / 0.1f;
        const float gz = spts[tid * 3 + 2] / 0.1f;
        const float fx = floorf(gx), fy = floorf(gy), fz = floorf(gz);
        sbase[tid][0] = (int)fx; sbase[tid][1] = (int)fy; sbase[tid][2] = (int)fz;
        sfrac[tid][0] = gx - fx; sfrac[tid][1] = gy - fy; sfrac[tid][2] = gz - fz;
    }
    __syncthreads();

    // ---- Phase 1b: one thread per (query, corner): hash + table gather ----
    {
        const int q = tid >> 3;
        const int c = tid & 7;
        // reference CORNERS order as bit codes: {0,1,2,4,3,5,6,7}
        int m = c;
        if (c == 3) m = 4; else if (c == 4) m = 3;
        const int cx = m & 1, cy = (m >> 1) & 1, cz = (m >> 2) & 1;
        const unsigned h = (((unsigned)(sbase[q][0] + cx)) * 73856093u
                          + ((unsigned)(sbase[q][1] + cy)) * 19349669u
                          + ((unsigned)(sbase[q][2] + cz)) * 83492791u) & TABLE_MASK;
        int v = table[h];
        const float wx = cx ? sfrac[q][0] : 1.0f - sfrac[q][0];
        const float wy = cy ? sfrac[q][1] : 1.0f - sfrac[q][1];
        const float wz = cz ? sfrac[q][2] : 1.0f - sfrac[q][2];
        float w = wx * wy * wz;
        if (v < 0) { v = 0; w = 0.0f; }   // empty slot: features[0] * 0
        sent[q][c] = make_int2(v, __float_as_int(w));
    }
    __syncthreads();

    // ---- Phase 2: one wave per query (4 queries per wave sequentially),
    //      lanes 0..29 each own one aligned float4 chunk of the 120-f row ----
    const int wv   = tid >> 5;
    const int lane = tid & 31;
    const int ch   = (lane < CH) ? lane : (CH - 1);   // lanes 30/31: harmless dup load
    const v4f* __restrict__ feat4 = (const v4f*)feat;
    v4f*       __restrict__ out4  = (v4f*)out;

    for (int i = 0; i < QPB / (NTHREADS / 32); ++i) {   // 4 iterations
        const int q  = wv * (QPB / (NTHREADS / 32)) + i;
        const int gq = qb + q;
        v4f acc = {0.0f, 0.0f, 0.0f, 0.0f};
        #pragma unroll
        for (int c = 0; c < 8; ++c) {
            const int2  e = sent[q][c];                 // ds_load_b64, broadcast
            const float w = __int_as_float(e.y);
            const v4f   f = feat4[(size_t)e.x * CH + ch]; // global_load_b128 (L2-hit)
            acc += f * w;
        }
        if (lane < CH && gq < M) {
            // Non-temporal: keep the 480 MB output stream out of L2 so the
            // 137 MB feature table stays resident.
            __builtin_nontemporal_store(acc, &out4[(size_t)gq * CH + ch]);
        }
    }
}

extern "C" void kernel_launch(void* const* d_in, const int* in_sizes, int n_in,
                              void* d_out, int out_size, void* d_ws, size_t ws_size,
                              hipStream_t stream) {
    const float* qp    = (const float*)d_in[0];   // [M,3] f32
    const float* feat  = (const float*)d_in[1];   // [n_vox,120] f32
    const int*   table = (const int*)d_in[2];     // [1<<20] i32
    float*       out   = (float*)d_out;           // [M,120] f32

    const int M      = in_sizes[0] / 3;
    const int blocks = (M + QPB - 1) / QPB;
    hipLaunchKernelGGL(voxel_trilinear_kernel, dim3(blocks), dim3(NTHREADS), 0, stream,
                       qp, feat, table, out, M);
}